// SE_HALF_6408091205884
// MI455X (gfx1250) — compile-verified
//
#include <hip/hip_runtime.h>
#include <hip/hip_bf16.h>

typedef __attribute__((ext_vector_type(2))) float v2f;
typedef __attribute__((ext_vector_type(8))) float v8f;

#define B   32
#define C   512
#define R   32      // C / ratio
#define HW  4096    // 64*64
#define HALF 256

// ---------------------------------------------------------------------------
// Kernel 1: global average pool. One wave (32 lanes) per (b,c) plane of 4096
// floats. float4 streaming loads, shfl-xor wave reduction (wave32).
// grid = 2048 blocks * 256 threads = 16384 waves = 32*512 planes.
// ---------------------------------------------------------------------------
__global__ void se_pool_kernel(const float* __restrict__ x,
                               float* __restrict__ pool) {
    const int lane  = threadIdx.x & 31;
    const int wave  = threadIdx.x >> 5;
    const int plane = blockIdx.x * 8 + wave;          // 0 .. 16383
    const float4* src = (const float4*)(x + (size_t)plane * HW);

    float s = 0.0f;
    #pragma unroll
    for (int i = 0; i < 32; ++i) {                    // 32 float4 per lane
        float4 v = src[lane + i * 32];
        s += v.x + v.y + v.z + v.w;
    }
    // wave32 reduction
    #pragma unroll
    for (int m = 16; m > 0; m >>= 1) s += __shfl_xor(s, m, 32);
    if (lane == 0) pool[plane] = s * (1.0f / (float)HW);
}

// ---------------------------------------------------------------------------
// Kernel 2: excitation MLP entirely in one 128-thread block (4 waves) using
// V_WMMA_F32_16X16X4_F32 (full f32 precision — ranking depends on exact
// gate values).
//   h = relu(y @ w1^T)        M=32 N=32  K=512   (4 waves -> one tile each)
//   g = sigmoid(h @ w2^T)     M=32 N=512 K=32    (64 tiles -> 16 per wave)
// Fragment layouts per ISA 7.12.2:
//   A (16x4):  lanes 0-15 M=lane,K={0,1}; lanes 16-31 M=lane-16,K={2,3}
//   B (4x16):  VGPR i holds rows K=i (lanes 0-15) / K=i+2 (lanes 16-31)
//   C/D:       VGPR i holds rows M=i (lanes 0-15) / M=i+8 (lanes 16-31)
// ---------------------------------------------------------------------------
__global__ void se_mlp_wmma_kernel(const float* __restrict__ pool,
                                   const float* __restrict__ w1,  // (32,512)
                                   const float* __restrict__ w2,  // (512,32)
                                   float* __restrict__ gates) {   // (32,512)
    __shared__ float hsm[32][32];

    const int lane = threadIdx.x & 31;
    const int wave = threadIdx.x >> 5;     // 0..3
    const int l16  = lane & 15;
    const int half = lane >> 4;            // 0: lanes 0-15, 1: lanes 16-31

    // ---- GEMM 1: h[32x32] = y[32x512] @ w1^T[512x32], relu -> LDS ----
    {
        const int mt = wave >> 1;          // M tile 0/1
        const int nt = wave & 1;           // N tile 0/1
        v8f acc = {};
        const int arow = mt * 16 + l16;    // batch row
        const int bn   = nt * 16 + l16;    // hidden unit (w1 row)
        for (int k0 = 0; k0 < C; k0 += 4) {
            v2f a, b;
            a.x = pool[arow * C + k0 + half * 2 + 0];
            a.y = pool[arow * C + k0 + half * 2 + 1];
            b.x = w1[bn * C + k0 + half * 2 + 0];
            b.y = w1[bn * C + k0 + half * 2 + 1];
            acc = __builtin_amdgcn_wmma_f32_16x16x4_f32(
                false, a, false, b, (short)0, acc, false, false);
        }
        #pragma unroll
        for (int i = 0; i < 8; ++i) {
            const int row = mt * 16 + i + half * 8;
            const int col = nt * 16 + l16;
            hsm[row][col] = fmaxf(acc[i], 0.0f);       // ReLU
        }
    }
    __syncthreads();

    // ---- GEMM 2: g[32x512] = h[32x32] @ w2^T[32x512], sigmoid ----
    for (int t = wave; t < 64; t += 4) {               // 16 tiles per wave
        const int mt = t & 1;                          // M tile 0/1
        const int nt = t >> 1;                         // N tile 0..31
        const int arow = mt * 16 + l16;
        const int bn   = nt * 16 + l16;                // channel (w2 row)
        v8f acc = {};
        #pragma unroll
        for (int ks = 0; ks < 8; ++ks) {
            const int k0 = ks * 4;
            v2f a, b;
            a.x = hsm[arow][k0 + half * 2 + 0];
            a.y = hsm[arow][k0 + half * 2 + 1];
            b.x = w2[bn * R + k0 + half * 2 + 0];
            b.y = w2[bn * R + k0 + half * 2 + 1];
            acc = __builtin_amdgcn_wmma_f32_16x16x4_f32(
                false, a, false, b, (short)0, acc, false, false);
        }
        #pragma unroll
        for (int i = 0; i < 8; ++i) {
            const int row = mt * 16 + i + half * 8;    // batch
            const int col = nt * 16 + l16;             // channel
            const float v = acc[i];
            gates[row * C + col] = 1.0f / (1.0f + __expf(-v));
        }
    }
}

// ---------------------------------------------------------------------------
// Kernel 3: per-batch top-256 by ranking (reproduces jax.lax.top_k descending
// order with stable ties -> lower index first). One block per batch.
// ---------------------------------------------------------------------------
__global__ void se_select_kernel(const float* __restrict__ gates,
                                 int* __restrict__ sel_idx,
                                 float* __restrict__ sel_gate) {
    __shared__ float g[C];
    const int b = blockIdx.x;
    const int c = threadIdx.x;
    g[c] = gates[b * C + c];
    __syncthreads();

    const float my = g[c];
    int rank = 0;
    #pragma unroll 8
    for (int j = 0; j < C; ++j) {
        const float o = g[j];
        rank += (o > my) || (o == my && j < c);
    }
    if (rank < HALF) {
        sel_idx[b * HALF + rank]  = c;
        sel_gate[b * HALF + rank] = my;
    }
}

// ---------------------------------------------------------------------------
// Kernel 4: gather selected channel planes and scale by gate. One block per
// (b, j) output plane; 256 threads * 4 float4 = 4096 floats.
// grid = 32*256 = 8192 blocks.
// ---------------------------------------------------------------------------
__global__ void se_gather_scale_kernel(const float* __restrict__ x,
                                       const int* __restrict__ sel_idx,
                                       const float* __restrict__ sel_gate,
                                       float* __restrict__ out) {
    const int b = blockIdx.x >> 8;         // / 256
    const int j = blockIdx.x & 255;        // % 256
    const int c = sel_idx[b * HALF + j];
    const float gate = sel_gate[b * HALF + j];

    const float4* src = (const float4*)(x   + ((size_t)b * C    + c) * HW);
    float4*       dst = (float4*)      (out + ((size_t)b * HALF + j) * HW);

    #pragma unroll
    for (int k = 0; k < 4; ++k) {
        const int idx = threadIdx.x + k * 256;         // 1024 float4 / plane
        float4 v = src[idx];
        v.x *= gate; v.y *= gate; v.z *= gate; v.w *= gate;
        dst[idx] = v;
    }
}

// ---------------------------------------------------------------------------
extern "C" void kernel_launch(void* const* d_in, const int* in_sizes, int n_in,
                              void* d_out, int out_size, void* d_ws, size_t ws_size,
                              hipStream_t stream) {
    const float* x  = (const float*)d_in[0];   // (32, 512, 64, 64)
    const float* w1 = (const float*)d_in[1];   // (32, 512)
    const float* w2 = (const float*)d_in[2];   // (512, 32)
    float* out = (float*)d_out;                // (32, 256, 64, 64)

    char* ws = (char*)d_ws;
    float* pool     = (float*)(ws);             // 32*512 f32  = 64 KB
    float* gates    = (float*)(ws + 65536);     // 32*512 f32  = 64 KB
    int*   sel_idx  = (int*)  (ws + 131072);    // 32*256 i32  = 32 KB
    float* sel_gate = (float*)(ws + 163840);    // 32*256 f32  = 32 KB

    se_pool_kernel<<<2048, 256, 0, stream>>>(x, pool);
    se_mlp_wmma_kernel<<<1, 128, 0, stream>>>(pool, w1, w2, gates);
    se_select_kernel<<<B, C, 0, stream>>>(gates, sel_idx, sel_gate);
    se_gather_scale_kernel<<<B * HALF, 256, 0, stream>>>(x, sel_idx, sel_gate, out);
}